// RoIPool_15418932592922
// MI455X (gfx1250) — compile-verified
//
#include <hip/hip_runtime.h>
#include <hip/hip_bf16.h>
#include <math.h>

// RoIPool forward for MI455X (gfx1250), async-LDS staged.
//
// input:  (B=2, C=256, H=64, W=64) f32  -- 8 MB, L2-resident (192 MB L2)
// rois:   (N, 5) f32 = (batch_idx, x1, y1, x2, y2)
// output: (N, C, 7, 7) f32              -- 51 MB streamed once -> NT stores
//
// Block = 128 threads = 4 wave32, one ROI x 2 channels:
//   stage 1: GLOBAL_LOAD_ASYNC_TO_LDS_B128 of full ROI rows (each wave-op
//            moves 2 contiguous 256B rows; ASYNCcnt-tracked)
//   stage 2: per-bin max reduction out of LDS (banked 4B access, no
//            cacheline amplification), coalesced non-temporal stores.

#define POOLED 7
#define SPATIAL_SCALE 0.0625f
#define CCONST 256
#define HCONST 64
#define WCONST 64
#define HW (HCONST * WCONST)
#define CH_PER_BLK 2
#define THREADS 128

#if defined(__has_builtin)
#if __has_builtin(__builtin_amdgcn_global_load_async_to_lds_b128)
#define HAVE_ASYNC_B128 1
#endif
#if __has_builtin(__builtin_amdgcn_global_load_async_to_lds_b32)
#define HAVE_ASYNC_B32 1
#endif
#if __has_builtin(__builtin_amdgcn_s_wait_asynccnt)
#define HAVE_WAIT_ASYNC 1
#endif
#endif

typedef int v4i __attribute__((vector_size(16)));            // 4 x i32
typedef __attribute__((address_space(1))) int* gas1_i32p;    // global int*
typedef __attribute__((address_space(3))) int* las3_i32p;    // LDS int*
typedef __attribute__((address_space(1))) v4i* gas1_v4ip;    // global v4i*
typedef __attribute__((address_space(3))) v4i* las3_v4ip;    // LDS v4i*

// Copy 16 bytes (4 floats), 16B-aligned on both sides.
__device__ __forceinline__ void stage_b128(const float* __restrict__ gp,
                                           float* __restrict__ lp) {
#if defined(HAVE_ASYNC_B128)
  __builtin_amdgcn_global_load_async_to_lds_b128(
      (gas1_v4ip)(void*)(float*)gp, (las3_v4ip)(void*)lp, /*offset=*/0,
      /*cpol=*/0);
#elif defined(HAVE_ASYNC_B32)
  // IOFFSET applies to BOTH the global and LDS address (CDNA5 ISA),
  // so a split b128 is four b32s with immediate offsets.
  gas1_i32p g = (gas1_i32p)(void*)(float*)gp;
  las3_i32p l = (las3_i32p)(void*)lp;
  __builtin_amdgcn_global_load_async_to_lds_b32(g, l, 0, 0);
  __builtin_amdgcn_global_load_async_to_lds_b32(g, l, 4, 0);
  __builtin_amdgcn_global_load_async_to_lds_b32(g, l, 8, 0);
  __builtin_amdgcn_global_load_async_to_lds_b32(g, l, 12, 0);
#else
  *(float4*)lp = *(const float4*)gp;  // synchronous fallback
#endif
}

__device__ __forceinline__ void stage_wait() {
#if defined(HAVE_ASYNC_B128) || defined(HAVE_ASYNC_B32)
#if defined(HAVE_WAIT_ASYNC)
  __builtin_amdgcn_s_wait_asynccnt(0);
#else
  asm volatile("s_wait_asynccnt 0x0" ::: "memory");
#endif
#endif
}

__global__ __launch_bounds__(THREADS) void roipool_fwd_lds_kernel(
    const float* __restrict__ x,     // (B, C, H, W)
    const float* __restrict__ rois,  // (N, 5)
    float* __restrict__ out) {       // (N, C, 7, 7)
  __shared__ float tile[CH_PER_BLK * HW];  // 32 KB

  const int n = blockIdx.x;                // ROI id (uniform)
  const int c0 = blockIdx.y * CH_PER_BLK;  // channel chunk base (uniform)
  const int tid = threadIdx.x;

  // ---- ROI descriptor: block-uniform -> scalar float ops (s_floor_f32...) ----
  const float* r = rois + (size_t)n * 5;
  const int b = (int)r[0];
  const int x1 = (int)floorf(r[1] * SPATIAL_SCALE + 0.5f);
  const int y1 = (int)floorf(r[2] * SPATIAL_SCALE + 0.5f);
  const int x2 = (int)floorf(r[3] * SPATIAL_SCALE + 0.5f);
  const int y2 = (int)floorf(r[4] * SPATIAL_SCALE + 0.5f);
  const int roi_wi = max(x2 - x1 + 1, 1);
  const int roi_hi = max(y2 - y1 + 1, 1);
  const float bin_w = (float)roi_wi / (float)POOLED;  // true f32 div (matches ref)
  const float bin_h = (float)roi_hi / (float)POOLED;

  // Row range touched by any bin = clipped ROI rows.
  const int rh0 = min(max(y1, 0), HCONST);
  const int rh1 = min(max(y1 + roi_hi, 0), HCONST);
  const int region_h = rh1 - rh0;  // 0..64

  // ---- Stage 1: async copy of full 64-wide rows [rh0, rh1) -> LDS ----
  // LDS layout: tile[cl*4096 + hl*64 + w] (absolute column, shifts only).
  // Lane map: row = tid>>4 (8 rows/pass), chunk = (tid&15)*4 floats.
  {
    const int row = tid >> 4;        // 0..7
    const int wq = (tid & 15) << 2;  // 0,4,...,60
    for (int cl = 0; cl < CH_PER_BLK; ++cl) {
      const float* __restrict__ src =
          x + ((size_t)b * CCONST + (size_t)(c0 + cl)) * HW;
      for (int h0 = 0; h0 < region_h; h0 += 8) {
        const int hl = h0 + row;
        if (hl < region_h) {
          const float* gp = src + (size_t)(rh0 + hl) * WCONST + wq;
          float* lp = &tile[cl * HW + hl * WCONST + wq];
          stage_b128(gp, lp);
        }
      }
    }
  }
  stage_wait();
  __syncthreads();

  // ---- Stage 2: bin max-reduction from LDS, NT store ----
  for (int j = tid; j < CH_PER_BLK * POOLED * POOLED; j += THREADS) {
    const int pw = j % POOLED;
    const int ph = (j / POOLED) % POOLED;
    const int cl = j / (POOLED * POOLED);

    int ws = min(max((int)floorf((float)pw * bin_w) + x1, 0), WCONST);
    int we = min(max((int)ceilf((float)(pw + 1) * bin_w) + x1, 0), WCONST);
    int hs = min(max((int)floorf((float)ph * bin_h) + y1, 0), HCONST);
    int he = min(max((int)ceilf((float)(ph + 1) * bin_h) + y1, 0), HCONST);

    float m = -INFINITY;
    for (int h = hs; h < he; ++h) {
      const float* __restrict__ lrow = &tile[cl * HW + (h - rh0) * WCONST];
      for (int w = ws; w < we; ++w) m = fmaxf(m, lrow[w]);
    }
    if (he <= hs || we <= ws) m = 0.0f;

    // (N, C, 7, 7): consecutive j -> consecutive addresses -> coalesced.
    __builtin_nontemporal_store(
        m, &out[((size_t)n * CCONST + (size_t)(c0 + cl)) * (POOLED * POOLED) +
                ph * POOLED + pw]);
  }
}

extern "C" void kernel_launch(void* const* d_in, const int* in_sizes, int n_in,
                              void* d_out, int out_size, void* d_ws,
                              size_t ws_size, hipStream_t stream) {
  const float* x = (const float*)d_in[0];     // (B, C, H, W) f32
  const float* rois = (const float*)d_in[1];  // (N, 5) f32
  const int N = in_sizes[1] / 5;

  dim3 grid((unsigned)N, (unsigned)(CCONST / CH_PER_BLK), 1);  // (1024, 128)
  dim3 block(THREADS, 1, 1);
  roipool_fwd_lds_kernel<<<grid, block, 0, stream>>>(x, rois, (float*)d_out);
}